// SparseAttention_86406152061640
// MI455X (gfx1250) — compile-verified
//
#include <hip/hip_runtime.h>

typedef float v2f __attribute__((ext_vector_type(2)));
typedef float v8f __attribute__((ext_vector_type(8)));

#define LSTRIDE 66   // 64 + 2 pad: bank-conflict-free row-strided ds_load_b64, 8B aligned

__global__ __launch_bounds__(128) void bdattn_kernel(const float* __restrict__ q,
                                                     const float* __restrict__ k,
                                                     const float* __restrict__ v,
                                                     float* __restrict__ out) {
    __shared__ float lds_k[64 * LSTRIDE];
    __shared__ float lds_v[64 * LSTRIDE];
    __shared__ float lds_s[64 * LSTRIDE];

    const int tid   = threadIdx.x;
    const int lane  = tid & 31;
    const int wave  = tid >> 5;       // 0..3
    const int l15   = lane & 15;
    const int hi    = lane >> 4;      // 0 or 1 (half-wave)
    const int h2    = hi * 2;         // K-offset for A/B operand layout
    const int mBase = wave * 16;      // this wave's 16 output rows

    // one 64x64 block per workgroup; blocks are contiguous 4096-float chunks
    const size_t base = (size_t)blockIdx.x * 4096;
    const float* kg = k + base;
    const float* vg = v + base;

    // ---- stage K and V tiles into LDS (coalesced v2f loads) ----
    #pragma unroll
    for (int i = 0; i < 16; ++i) {
        int idx = tid + i * 128;          // v2f index 0..2047
        int row = idx >> 5;               // 32 v2f per 64-float row
        int col = (idx & 31) * 2;
        *(v2f*)&lds_k[row * LSTRIDE + col] = ((const v2f*)kg)[idx];
        *(v2f*)&lds_v[row * LSTRIDE + col] = ((const v2f*)vg)[idx];
    }

    // ---- preload this wave's Q slab directly in WMMA A-operand layout ----
    // lane L<16: rows mBase+L, K pair (kb*4+0, kb*4+1); lane L+16: K pair (+2,+3)
    const float* qg = q + base + (size_t)(mBase + l15) * 64 + h2;
    v2f Aq[16];
    #pragma unroll
    for (int kb = 0; kb < 16; ++kb) Aq[kb] = *(const v2f*)(qg + kb * 4);

    __syncthreads();

    // ---- S = Q * K^T  (f32 WMMA 16x16x4, accumulate over d=64 in 16 steps) ----
    v8f accS[4] = {};
    #pragma unroll
    for (int t = 0; t < 4; ++t) {
        // B = K^T: B[d][n] = K[n][d]; identical per-lane addressing to A-pattern
        const float* kp = &lds_k[(t * 16 + l15) * LSTRIDE + h2];
        #pragma unroll
        for (int kb = 0; kb < 16; ++kb) {
            v2f Bk = *(const v2f*)(kp + kb * 4);
            accS[t] = __builtin_amdgcn_wmma_f32_16x16x4_f32(
                false, Aq[kb], false, Bk, (short)0, accS[t], false, false);
        }
    }

    // ---- spill scores to LDS (re-layout C/D -> A for the second GEMM) ----
    {
        const int rowOff = (mBase + hi * 8) * LSTRIDE + l15;
        #pragma unroll
        for (int t = 0; t < 4; ++t)
            #pragma unroll
            for (int r = 0; r < 8; ++r)
                lds_s[rowOff + r * LSTRIDE + t * 16] = accS[t][r];
    }
    __syncthreads();

    // ---- softmax, one row per thread (SCALE = 1/sqrt(64) fused into exp) ----
    if (tid < 64) {
        float* row = &lds_s[tid * LSTRIDE];
        float m = row[0];
        #pragma unroll 8
        for (int j = 1; j < 64; ++j) m = fmaxf(m, row[j]);
        float s = 0.0f;
        #pragma unroll 8
        for (int j = 0; j < 64; ++j) {
            float e = __expf((row[j] - m) * 0.125f);
            row[j] = e;
            s += e;
        }
        float inv = 1.0f / s;
        #pragma unroll 8
        for (int j = 0; j < 64; ++j) row[j] *= inv;
    }
    __syncthreads();

    // ---- O = P * V ----
    v8f accO[4] = {};
    const float* pp  = &lds_s[(mBase + l15) * LSTRIDE + h2];   // A-pattern on P
    const float* vp0 = &lds_v[h2 * LSTRIDE + l15];             // B-pattern on V
    #pragma unroll
    for (int kb = 0; kb < 16; ++kb) {
        v2f Ap = *(const v2f*)(pp + kb * 4);
        const float* vk = vp0 + (kb * 4) * LSTRIDE;
        #pragma unroll
        for (int t = 0; t < 4; ++t) {
            v2f Bv;
            Bv.x = vk[t * 16];
            Bv.y = vk[t * 16 + LSTRIDE];
            accO[t] = __builtin_amdgcn_wmma_f32_16x16x4_f32(
                false, Ap, false, Bv, (short)0, accO[t], false, false);
        }
    }

    // ---- store O from C/D layout straight to global ----
    float* og = out + base + (size_t)(mBase + hi * 8) * 64 + l15;
    #pragma unroll
    for (int t = 0; t < 4; ++t)
        #pragma unroll
        for (int r = 0; r < 8; ++r)
            og[r * 64 + t * 16] = accO[t][r];
}

extern "C" void kernel_launch(void* const* d_in, const int* in_sizes, int n_in,
                              void* d_out, int out_size, void* d_ws, size_t ws_size,
                              hipStream_t stream) {
    const float* q = (const float*)d_in[0];
    const float* k = (const float*)d_in[1];
    const float* v = (const float*)d_in[2];
    float* out = (float*)d_out;

    // total elements / (64*64) = number of independent 64x64 blocks
    int nblocks = in_sizes[0] / 4096;   // 4*16*4096*64 / 4096 = 4096
    bdattn_kernel<<<dim3(nblocks), dim3(128), 0, stream>>>(q, k, v, out);
}